// EdgeToNodeEmbedding_7387343749430
// MI455X (gfx1250) — compile-verified
//
#include <hip/hip_runtime.h>
#include <hip/hip_bf16.h>

typedef float v2f __attribute__((ext_vector_type(2)));
typedef float v4f __attribute__((ext_vector_type(4)));
typedef float v8f __attribute__((ext_vector_type(8)));

#define EMB 128
#define NFEAT 128
#define KTOT 256

// Guaranteed-native f32 global atomic add (non-returning -> STOREcnt).
// CDNA5 ISA 15.18.3 opcode 86: GLOBAL_ATOMIC_ADD_F32, GV addressing mode.
__device__ __forceinline__ void atomic_add_f32_global(float* p, float v) {
    asm volatile("global_atomic_add_f32 %0, %1, off"
                 :: "v"((unsigned long long)(uintptr_t)p), "v"(v)
                 : "memory");
}

// ---------------------------------------------------------------------------
// Kernel 1: zero the h_aggr accumulator (lives in d_ws)
// ---------------------------------------------------------------------------
__global__ __launch_bounds__(256) void zero_f32(float* __restrict__ p, int n) {
    int i = blockIdx.x * 256 + threadIdx.x;
    if (i < n) p[i] = 0.0f;
}

// ---------------------------------------------------------------------------
// Kernel 2: scatter-add edge embeddings into destination nodes.
// One wave (32 lanes) per edge; each lane owns 4 consecutive floats.
// h is streamed once -> non-temporal b128 loads so L2 stays free for h_aggr
// (25.6 MB, resident in the 192 MB L2), which absorbs the f32 atomics.
// ---------------------------------------------------------------------------
__global__ __launch_bounds__(256) void scatter_add(
    const float* __restrict__ h, const int* __restrict__ edge_dst,
    float* __restrict__ hagg, int E)
{
    int gwave = (blockIdx.x * 256 + threadIdx.x) >> 5;   // edge id
    int lane  = threadIdx.x & 31;
    if (gwave >= E) return;

    const int dst = edge_dst[gwave];
    const v4f v = __builtin_nontemporal_load(
        (const v4f*)(h + (size_t)gwave * EMB + lane * 4));

    float* p = hagg + (size_t)dst * EMB + lane * 4;
    atomic_add_f32_global(p + 0, v.x);
    atomic_add_f32_global(p + 1, v.y);
    atomic_add_f32_global(p + 2, v.z);
    atomic_add_f32_global(p + 3, v.w);
}

// ---------------------------------------------------------------------------
// Kernel 3: out = relu([x | h_aggr] @ W^T + b) via V_WMMA_F32_16X16X4_F32.
// Block = 256 threads = 8 waves. blockIdx.x -> 16-row tile, wave -> 16-col tile
// (8 waves cover all 128 output columns). Each wave iterates K=256 in steps
// of 4. The concat boundary (k=128) is a multiple of the K-step, so the A
// source (x vs h_aggr) is wave-uniform -> EXEC stays all-1s around WMMA.
//
// f32 16x16x4 fragment layout (ISA 7.12.2):
//   A (16x4): lanes 0-15 hold M=lane, VGPR0/1 = K=k0,k0+1; lanes 16-31 K=k0+2,k0+3
//   B (4x16): lanes 0-15 hold N=lane, VGPR0/1 = K=k0,k0+1; lanes 16-31 K=k0+2,k0+3
//   D (16x16): row = r + 8*(lane>=16), col = lane&15
// ---------------------------------------------------------------------------
__global__ __launch_bounds__(256) void fused_gemm_relu(
    const float* __restrict__ x, const float* __restrict__ hagg,
    const float* __restrict__ W, const float* __restrict__ b,
    float* __restrict__ out, int N)
{
    const int wave = threadIdx.x >> 5;
    const int lane = threadIdx.x & 31;
    const int half = lane >> 4;          // 0: K pair {0,1}, 1: K pair {2,3}
    const int l15  = lane & 15;

    const int row0 = blockIdx.x * 16;
    const int col0 = wave * 16;

    int arow = row0 + l15;
    if (arow >= N) arow = N - 1;         // clamp loads; stores are guarded
    const int bcol = col0 + l15;         // W row index (output feature)

    const float* xrow = x    + (size_t)arow * NFEAT;
    const float* grow = hagg + (size_t)arow * EMB;
    const float* wrow = W    + (size_t)bcol * KTOT;

    v8f acc = {};

    // K = 0..127 : A from x
    #pragma unroll 8
    for (int k0 = 0; k0 < NFEAT; k0 += 4) {
        const int ka = k0 + 2 * half;
        v2f a  = *(const v2f*)(xrow + ka);
        v2f bb = *(const v2f*)(wrow + ka);
        acc = __builtin_amdgcn_wmma_f32_16x16x4_f32(
            false, a, false, bb, (short)0, acc, false, false);
    }
    // K = 128..255 : A from h_aggr
    #pragma unroll 8
    for (int k0 = 0; k0 < EMB; k0 += 4) {
        const int ka = k0 + 2 * half;
        v2f a  = *(const v2f*)(grow + ka);
        v2f bb = *(const v2f*)(wrow + NFEAT + ka);
        acc = __builtin_amdgcn_wmma_f32_16x16x4_f32(
            false, a, false, bb, (short)0, acc, false, false);
    }

    // Epilogue: bias + relu, scatter D fragment to out[N, 128]
    const float bias = b[col0 + l15];
    const int rbase = row0 + 8 * half;
    float* ocol = out + (size_t)rbase * EMB + col0 + l15;
    #pragma unroll
    for (int r = 0; r < 8; ++r) {
        if (rbase + r < N) {
            float v = acc[r] + bias;
            ocol[(size_t)r * EMB] = v > 0.0f ? v : 0.0f;
        }
    }
}

// ---------------------------------------------------------------------------
// Host-side launcher. Inputs: h[E,128], x[N,128], edge_dst[E] (i32),
// W[128,256], b[128]. Output: [N,128] f32. d_ws holds h_aggr (N*128*4 B).
// ---------------------------------------------------------------------------
extern "C" void kernel_launch(void* const* d_in, const int* in_sizes, int n_in,
                              void* d_out, int out_size, void* d_ws, size_t ws_size,
                              hipStream_t stream) {
    const float* h        = (const float*)d_in[0];
    const float* x        = (const float*)d_in[1];
    const int*   edge_dst = (const int*)  d_in[2];
    const float* W        = (const float*)d_in[3];
    const float* b        = (const float*)d_in[4];
    float*       out      = (float*)d_out;
    float*       hagg     = (float*)d_ws;          // N*EMB floats (25.6 MB)

    const int E = in_sizes[0] / EMB;               // 800000
    const int N = in_sizes[1] / NFEAT;             // 50000

    // 1) zero accumulator
    {
        int n = N * EMB;
        zero_f32<<<(n + 255) / 256, 256, 0, stream>>>(hagg, n);
    }
    // 2) scatter-add (one wave per edge)
    {
        long long threads = (long long)E * 32;
        int blocks = (int)((threads + 255) / 256);
        scatter_add<<<blocks, 256, 0, stream>>>(h, edge_dst, hagg, E);
    }
    // 3) fused concat-GEMM + bias + relu (WMMA f32)
    {
        int row_tiles = (N + 15) / 16;
        fused_gemm_relu<<<row_tiles, 256, 0, stream>>>(x, hagg, W, b, out, N);
    }
}